// WeightedVarianceCrossEntropyLoss_14869176779175
// MI455X (gfx1250) — compile-verified
//
#include <hip/hip_runtime.h>
#include <hip/hip_bf16.h>
#include <stdint.h>

#define EPS_F      1e-7f
#define LOG_EPS    -16.11809565095832f      // log(1e-7)
#define LOG_1M_EPS -1.0000000116861e-7f     // log(1 - 1e-7)

typedef __attribute__((ext_vector_type(2))) float v2f;
typedef __attribute__((ext_vector_type(8))) float v8f;
typedef __attribute__((ext_vector_type(4))) int   v4i;

typedef __attribute__((address_space(1))) v4i* as1_v4i;
typedef __attribute__((address_space(3))) v4i* as3_v4i;

// ---- CDNA5 async global->LDS copy (ASYNCcnt-tracked DMA path) ----
__device__ __forceinline__ void async_cp_b128(const float* gsrc, float* lds_dst) {
#if __has_builtin(__builtin_amdgcn_global_load_async_to_lds_b128)
  __builtin_amdgcn_global_load_async_to_lds_b128(
      (as1_v4i)(void*)gsrc, (as3_v4i)lds_dst, /*offset=*/0, /*cpol=*/0);
#else
  unsigned lds_off = (unsigned)(uintptr_t)(__attribute__((address_space(3))) void*)lds_dst;
  asm volatile("global_load_async_to_lds_b128 %0, %1, off"
               :: "v"(lds_off), "v"(gsrc) : "memory");
#endif
}

__device__ __forceinline__ void wait_async0() {
#if __has_builtin(__builtin_amdgcn_s_wait_asynccnt)
  __builtin_amdgcn_s_wait_asynccnt(0);
  asm volatile("" ::: "memory");
#else
  asm volatile("s_wait_asynccnt 0" ::: "memory");
#endif
}

// ---- per-element weighted softmax-BCE term ----
__device__ __forceinline__ float bce_term(float x, float logZ, float t) {
  float lp = x - logZ;                  // log softmax (exact)
  float p  = __expf(lp);
  p = fminf(fmaxf(p, EPS_F), 1.0f - EPS_F);
  float logp   = fminf(fmaxf(lp, LOG_EPS), LOG_1M_EPS);   // clip in log domain: no extra log
  float log1mp = __logf(1.0f - p);
  return -(t * logp + (1.0f - t) * log1mp);
}

// One workgroup per row. 256 threads x 16 elements = 4096 columns.
// Interleaved ownership: thread (wave,lane) handles floats
//   wave*512 + s*128 + lane*4 .. +3   for s = 0..3
// so every wave-level b128 access (global load, async DMA, LDS read) is a
// single contiguous 512B burst.
__global__ __launch_bounds__(256) void wbce_row_kernel(
    const float* __restrict__ y_pred, const float* __restrict__ y_true,
    const float* __restrict__ weights, const long long* __restrict__ cond,
    float* __restrict__ partial, float inv_bc) {
  const int row  = blockIdx.x;
  const int tid  = threadIdx.x;
  const int lane = tid & 31;
  const int wave = tid >> 5;
  const int base = wave * 512 + lane * 4;

  __shared__ float yt[4096];   // staged y_true row (16 KB)
  __shared__ float red[8];

  const float* pr = y_pred + (size_t)row * 4096;
  const float* tr = y_true + (size_t)row * 4096;

  // Kick off async DMA of this wave's y_true slice into LDS (overlaps pass 1)
#pragma unroll
  for (int s = 0; s < 4; ++s)
    async_cp_b128(tr + base + s * 128, &yt[base + s * 128]);

  // Row weight (uniform -> scalar path)
  float w = weights[(int)cond[row]];

  // y_pred row -> registers (read from HBM exactly once, fully coalesced)
  float4 x[4];
#pragma unroll
  for (int s = 0; s < 4; ++s) x[s] = *(const float4*)(pr + base + s * 128);

  // ---- pass 1a: row max ----
  float m = -INFINITY;
#pragma unroll
  for (int s = 0; s < 4; ++s)
    m = fmaxf(m, fmaxf(fmaxf(x[s].x, x[s].y), fmaxf(x[s].z, x[s].w)));
#pragma unroll
  for (int off = 16; off; off >>= 1) m = fmaxf(m, __shfl_xor(m, off, 32));
  if (lane == 0) red[wave] = m;
  __syncthreads();
  float rowmax = red[0];
#pragma unroll
  for (int i = 1; i < 8; ++i) rowmax = fmaxf(rowmax, red[i]);
  __syncthreads();

  // ---- pass 1b: sum of exp ----
  float se = 0.0f;
#pragma unroll
  for (int s = 0; s < 4; ++s) {
    se += __expf(x[s].x - rowmax);
    se += __expf(x[s].y - rowmax);
    se += __expf(x[s].z - rowmax);
    se += __expf(x[s].w - rowmax);
  }
#pragma unroll
  for (int off = 16; off; off >>= 1) se += __shfl_xor(se, off, 32);
  if (lane == 0) red[wave] = se;
  __syncthreads();
  float S = 0.0f;
#pragma unroll
  for (int i = 0; i < 8; ++i) S += red[i];
  float logZ = rowmax + __logf(S);
  __syncthreads();

  // ---- pass 2: BCE against LDS-staged y_true ----
  wait_async0();   // this wave's async copies are done (we only read our own slice)
  float acc = 0.0f;
#pragma unroll
  for (int s = 0; s < 4; ++s) {
    float4 t4 = *(const float4*)&yt[base + s * 128];
    float4 xv = x[s];
    acc += bce_term(xv.x, logZ, t4.x);
    acc += bce_term(xv.y, logZ, t4.y);
    acc += bce_term(xv.z, logZ, t4.z);
    acc += bce_term(xv.w, logZ, t4.w);
  }
#pragma unroll
  for (int off = 16; off; off >>= 1) acc += __shfl_xor(acc, off, 32);
  if (lane == 0) red[wave] = acc;
  __syncthreads();
  if (tid == 0) {
    float bs = 0.0f;
#pragma unroll
    for (int i = 0; i < 8; ++i) bs += red[i];
    partial[row] = bs * w * inv_bc;
  }
}

// Final reduction of 16384 per-row partials, single wave32.
// Uses V_WMMA_F32_16X16X4_F32 with A = ones as a 4-way f32 adder tree:
// D[m,n] = sum_k B[k,n] + C[m,n]  -> accumulated column sums, then 16-lane shuffle.
__global__ __launch_bounds__(32) void final_reduce_kernel(
    const float* __restrict__ partial, float* __restrict__ out, int n) {
  const int lane = threadIdx.x;   // 0..31, full wave, EXEC all ones
  v8f c = {0.f, 0.f, 0.f, 0.f, 0.f, 0.f, 0.f, 0.f};
  v2f a; a[0] = 1.0f; a[1] = 1.0f;
  for (int i = 0; i < n; i += 64) {
    v2f b;
    b[0] = partial[i + lane];
    b[1] = partial[i + 32 + lane];
    c = __builtin_amdgcn_wmma_f32_16x16x4_f32(
        /*neg_a=*/false, a, /*neg_b=*/false, b,
        /*c_mod=*/(short)0, c, /*reuse_a=*/false, /*reuse_b=*/false);
  }
  // c[0] at lane L holds column-sum (lanes 16..31 duplicate rows M=8): reduce 16 lanes
  float s = c[0];
#pragma unroll
  for (int off = 8; off; off >>= 1) s += __shfl_xor(s, off, 32);
  if (lane == 0) out[0] = s;
}

extern "C" void kernel_launch(void* const* d_in, const int* in_sizes, int n_in,
                              void* d_out, int out_size, void* d_ws, size_t ws_size,
                              hipStream_t stream) {
  const float*     y_pred  = (const float*)d_in[0];
  const float*     y_true  = (const float*)d_in[1];
  const float*     weights = (const float*)d_in[2];
  const long long* cond    = (const long long*)d_in[3];
  float* out     = (float*)d_out;
  float* partial = (float*)d_ws;

  const int B = in_sizes[3];              // 16384
  const int C = in_sizes[0] / B;          // 4096
  const float inv_bc = 1.0f / ((float)B * (float)C);

  wbce_row_kernel<<<B, 256, 0, stream>>>(y_pred, y_true, weights, cond, partial, inv_bc);
  final_reduce_kernel<<<1, 32, 0, stream>>>(partial, out, B);
}